// SelectiveScan1d_65171833749968
// MI455X (gfx1250) — compile-verified
//
#include <hip/hip_runtime.h>
#include <hip/hip_fp16.h>
#include <cstdint>

#define CH        64
#define ST        16
#define LL        512
#define NTHREADS  512
#define PROJ_ROWS 36
#define PLL       516   // padded proj row stride (floats): 516 % 64 = 4 -> conflict free
#define WLD       68    // padded weight row stride (floats): 68 % 64 = 4 -> conflict free

// 1 = issue x staging via async global->LDS DMA (inline asm, ASYNCcnt tracked)
#define USE_ASYNC_ASM 1

typedef float v2f __attribute__((ext_vector_type(2)));
typedef float v8f __attribute__((ext_vector_type(8)));

__device__ __forceinline__ float softplusf(float z) {
  // stable: max(z,0) + log(1 + exp(-|z|))
  return fmaxf(z, 0.f) + __logf(1.f + __expf(-fabsf(z)));
}

__global__ void __launch_bounds__(NTHREADS, 1)
selective_scan_fused(const float* __restrict__ x,
                     const float* __restrict__ Wx,
                     const float* __restrict__ Wdt,
                     const float* __restrict__ b_dt,
                     const float* __restrict__ A_log,
                     const float* __restrict__ Dvec,
                     const float* __restrict__ Wg,
                     const float* __restrict__ bg,
                     float* __restrict__ out)
{
  __shared__ float    x_lds[CH * LL];              // 128 KB  (x[b], [c][l])
  __shared__ float    proj_lds[PROJ_ROWS * PLL];   // ~72.6 KB (dt_lr | B_t | C_t)
  __shared__ _Float16 gate_lds[CH * LL];           // 64 KB   sigmoid(gate) in f16
  __shared__ float    wx_lds[48 * WLD];            // 12.75 KB, rows 36..47 zero
  __shared__ float    wg_lds[64 * WLD];            // 17 KB

  const int t    = threadIdx.x;
  const int b    = blockIdx.x;
  const int lane = t & 31;
  const int wave = t >> 5;
  const int ln   = lane & 15;   // N / M-row selector within half-wave
  const int hi   = lane >> 4;   // which half of the wave

  // -------- stage x[b] (64x512 f32) + weights into LDS --------
  const float4* xg  = (const float4*)(x + (size_t)b * CH * LL);
  float4*       xl4 = (float4*)x_lds;

#if USE_ASYNC_ASM
  for (int i = t; i < CH * LL / 4; i += NTHREADS) {
    // LDS byte offset (AS3 pointer value == LDS offset) per lane
    unsigned ldsoff =
        (unsigned)(uintptr_t)(__attribute__((address_space(3))) void*)(xl4 + i);
    unsigned long long gaddr = (unsigned long long)(xg + i);
    asm volatile("global_load_async_to_lds_b128 %0, %1, off"
                 :
                 : "v"(ldsoff), "v"(gaddr)
                 : "memory");
  }
#else
  for (int i = t; i < CH * LL / 4; i += NTHREADS) xl4[i] = xg[i];
#endif

  for (int i = t; i < 48 * WLD; i += NTHREADS) {
    int r = i / WLD, c = i - r * WLD;
    wx_lds[i] = (r < PROJ_ROWS && c < CH) ? Wx[r * CH + c] : 0.f;
  }
  for (int i = t; i < 64 * WLD; i += NTHREADS) {
    int r = i / WLD, c = i - r * WLD;
    wg_lds[i] = (c < CH) ? Wg[r * CH + c] : 0.f;
  }

#if USE_ASYNC_ASM
  asm volatile("s_wait_asynccnt 0x0" ::: "memory");
#endif
  __syncthreads();

  // -------- WMMA phase: proj = Wx*x (3 M-tiles) and gate = Wg*x (4 M-tiles),
  //          32 N-tiles of L each -> 224 tiles round-robined over 16 waves ----
  for (int tile = wave; tile < 224; tile += 16) {
    const int  nt     = tile & 31;
    const int  sel    = tile >> 5;          // 0..2 proj, 3..6 gate
    const int  l0     = nt * 16;
    const bool isproj = (sel < 3);
    const int  m0     = isproj ? sel * 16 : (sel - 3) * 16;
    const float* W    = isproj ? wx_lds : wg_lds;

    v8f acc = {};
#pragma unroll
    for (int k = 0; k < CH; k += 4) {
      const int kk = k + hi * 2;            // K pair per ISA A-layout (16x4 f32)
      v2f a, bfr;
      a.x   = W[(m0 + ln) * WLD + kk];
      a.y   = W[(m0 + ln) * WLD + kk + 1];
      bfr.x = x_lds[kk * LL + l0 + ln];     // B[k, N] with N = ln (contiguous)
      bfr.y = x_lds[(kk + 1) * LL + l0 + ln];
      acc = __builtin_amdgcn_wmma_f32_16x16x4_f32(false, a, false, bfr,
                                                  (short)0, acc, false, false);
    }
    const int mrow = hi * 8;                // C/D layout: VGPR v -> M = v + 8*hi
    if (isproj) {
#pragma unroll
      for (int v = 0; v < 8; ++v) {
        int M = m0 + mrow + v;
        if (M < PROJ_ROWS) proj_lds[M * PLL + l0 + ln] = acc[v];
      }
    } else {
#pragma unroll
      for (int v = 0; v < 8; ++v) {
        int M = m0 + mrow + v;
        float g = acc[v] + bg[M];
        gate_lds[M * LL + l0 + ln] = (_Float16)(1.f / (1.f + __expf(-g)));
      }
    }
  }
  __syncthreads();

  // -------- scan phase: thread -> (state n = t&15, channels c0 = t>>4, c1 = c0+32)
  const int n  = t & 15;
  const int cg = t >> 4;          // 0..31
  const int c0 = cg;
  const int c1 = cg + 32;

  float wdt0[4], wdt1[4];
#pragma unroll
  for (int r = 0; r < 4; ++r) { wdt0[r] = Wdt[c0 * 4 + r]; wdt1[r] = Wdt[c1 * 4 + r]; }
  const float bdt0 = b_dt[c0], bdt1 = b_dt[c1];
  const float An0  = -__expf(A_log[c0 * ST + n]);
  const float An1  = -__expf(A_log[c1 * ST + n]);
  const float D0   = Dvec[c0], D1 = Dvec[c1];

  float h0 = 0.f, h1 = 0.f;
  float stash0 = 0.f, stash1 = 0.f;
  float* outb = out + (size_t)b * CH * LL;

  for (int l = 0; l < LL; ++l) {
    // dt inputs (broadcast reads within each 16-lane half)
    const float p0 = proj_lds[0 * PLL + l];
    const float p1 = proj_lds[1 * PLL + l];
    const float p2 = proj_lds[2 * PLL + l];
    const float p3 = proj_lds[3 * PLL + l];
    const float Btl = proj_lds[(4  + n) * PLL + l];
    const float Ctl = proj_lds[(20 + n) * PLL + l];

    // channel c0
    const float xl0 = x_lds[c0 * LL + l];
    float z0  = wdt0[0]*p0 + wdt0[1]*p1 + wdt0[2]*p2 + wdt0[3]*p3 + bdt0;
    float dt0 = softplusf(z0);
    float dA0 = __expf(dt0 * An0);
    h0 = dA0 * h0 + dt0 * Btl * xl0;
    float y0 = h0 * Ctl;

    // channel c1
    const float xl1 = x_lds[c1 * LL + l];
    float z1  = wdt1[0]*p0 + wdt1[1]*p1 + wdt1[2]*p2 + wdt1[3]*p3 + bdt1;
    float dt1 = softplusf(z1);
    float dA1 = __expf(dt1 * An1);
    h1 = dA1 * h1 + dt1 * Btl * xl1;
    float y1 = h1 * Ctl;

    // reduce over the 16 states (width-16 xor reduce within half-wave)
#pragma unroll
    for (int m = 8; m >= 1; m >>= 1) {
      y0 += __shfl_xor(y0, m, 16);
      y1 += __shfl_xor(y1, m, 16);
    }

    const float g0 = (float)gate_lds[c0 * LL + l];
    const float g1 = (float)gate_lds[c1 * LL + l];
    const float v0 = (y0 + xl0 * D0) * g0;
    const float v1 = (y1 + xl1 * D1) * g1;

    // stage 16 results, then do one coalesced 16-wide store per channel row
    if (n == (l & 15)) { stash0 = v0; stash1 = v1; }
    if ((l & 15) == 15) {
      const int lb = l & ~15;
      outb[c0 * LL + lb + n] = stash0;
      outb[c1 * LL + lb + n] = stash1;
    }
  }
}

extern "C" void kernel_launch(void* const* d_in, const int* in_sizes, int n_in,
                              void* d_out, int out_size, void* d_ws, size_t ws_size,
                              hipStream_t stream) {
  const float* x     = (const float*)d_in[0];
  const float* Wx    = (const float*)d_in[1];
  const float* Wdt   = (const float*)d_in[2];
  const float* b_dt  = (const float*)d_in[3];
  const float* A_log = (const float*)d_in[4];
  const float* D     = (const float*)d_in[5];
  const float* Wg    = (const float*)d_in[6];
  const float* bg    = (const float*)d_in[7];
  float* out = (float*)d_out;

  const int B = in_sizes[0] / (CH * LL);   // 128
  hipLaunchKernelGGL(selective_scan_fused, dim3(B), dim3(NTHREADS), 0, stream,
                     x, Wx, Wdt, b_dt, A_log, D, Wg, bg, out);
}